// JacobianDetLoss_20237885899194
// MI455X (gfx1250) — compile-verified
//
#include <hip/hip_runtime.h>

typedef float f4  __attribute__((ext_vector_type(4)));
typedef float v2f __attribute__((ext_vector_type(2)));
typedef float v8f __attribute__((ext_vector_type(8)));

namespace {
constexpr int      S    = 192;
constexpr int      SI   = 191;               // interior extent per dim
constexpr int      SS   = S * S;             // 36864
constexpr int      SSS  = S * S * S;         // 7077888
constexpr int      ZB   = (SI + 3) / 4;      // 48 z-quads per row
constexpr unsigned UNITS = 2u * SI * SI * ZB;    // (b,x,y,zquad) work units
constexpr unsigned NM4   = (2u * (unsigned)SSS) / 4; // mask elements as float4
}

// 32-lane sum via V_WMMA_F32_16X16X4_F32.
// A (16x4): VGPR0 = lanes 0..15 -> (M=lane,K=0), lanes 16..31 -> (M=lane-16,K=2);
// with a.y = 0 the row-sum of row m is acc[m] + acc[m+16].
// B = all-ones => D[m,n] = rowsum(m).  D VGPR r: lanes 0..15 hold M=r,
// lanes 16..31 hold M=r+8, so summing d[0..7] gives half-wave sums; one
// xor-16 shuffle completes the full 32-lane sum.  EXEC must be all 1s here.
__device__ __forceinline__ float wave_sum_wmma(float v) {
    v2f a; a.x = v;    a.y = 0.0f;
    v2f b; b.x = 1.0f; b.y = 1.0f;
    v8f c = {};
    c = __builtin_amdgcn_wmma_f32_16x16x4_f32(false, a, false, b,
                                              (short)0, c, false, false);
    float s = c[0] + c[1] + c[2] + c[3] + c[4] + c[5] + c[6] + c[7];
    s += __shfl_xor(s, 16, 32);
    return s;
}

__global__ void jdet_init_ws(float* ws) {
    ws[0] = 0.0f;
    ws[1] = 0.0f;
}

__global__ void jdet_main(const float* __restrict__ disp,
                          const float* __restrict__ mask,
                          float* __restrict__ ws) {
    const unsigned tid    = blockIdx.x * blockDim.x + threadIdx.x;
    const unsigned stride = gridDim.x * blockDim.x;

    const f4 zf = (f4)(0.0f);

    // ---- numerator: sum over interior voxels of relu(-det) * mask ----
    f4 accNv = zf;
    for (unsigned u = tid; u < UNITS; u += stride) {
        unsigned zb = u % ZB;  unsigned t = u / ZB;
        unsigned y  = t % SI;  t /= SI;
        unsigned x  = t % SI;  unsigned bb = t / SI;

        const int z0  = (int)zb * 4;
        const int idx = (int)x * SS + (int)y * S + z0;  // 16B aligned

        // d channel reversal: d ch0 = disp ch2, d ch1 = disp ch1, d ch2 = disp ch0
        const float* p0 = disp + ((size_t)bb * 3 + 2) * (size_t)SSS;
        const float* p1 = disp + ((size_t)bb * 3 + 1) * (size_t)SSS;
        const float* p2 = disp + ((size_t)bb * 3 + 0) * (size_t)SSS;

        // corner rows (z0..z0+3) + one scalar for the z-shift (all in-bounds)
        f4 c0 = *(const f4*)(p0 + idx);
        f4 c1 = *(const f4*)(p1 + idx);
        f4 c2 = *(const f4*)(p2 + idx);
        float s0 = p0[idx + 4], s1 = p1[idx + 4], s2 = p2[idx + 4];

        // x+1 rows
        f4 x0 = *(const f4*)(p0 + idx + SS);
        f4 x1 = *(const f4*)(p1 + idx + SS);
        f4 x2 = *(const f4*)(p2 + idx + SS);
        // y+1 rows
        f4 y0 = *(const f4*)(p0 + idx + S);
        f4 y1 = *(const f4*)(p1 + idx + S);
        f4 y2 = *(const f4*)(p2 + idx + S);
        // z+1 values: shift corner row left by one, append scalar
        f4 z0v = { c0[1], c0[2], c0[3], s0 };
        f4 z1v = { c1[1], c1[2], c1[3], s1 };
        f4 z2v = { c2[1], c2[2], c2[3], s2 };

        f4 a00 = (x0 - c0) + 1.0f;
        f4 a01 =  x1 - c1;
        f4 a02 =  x2 - c2;
        f4 a10 =  y0 - c0;
        f4 a11 = (y1 - c1) + 1.0f;
        f4 a12 =  y2 - c2;
        f4 a20 =  z0v - c0;
        f4 a21 =  z1v - c1;
        f4 a22 = (z2v - c2) + 1.0f;

        f4 det = a00 * (a11 * a22 - a12 * a21)
               + a01 * (a12 * a20 - a10 * a22)
               + a02 * (a10 * a21 - a11 * a20);

        f4 pen = __builtin_elementwise_max(-det, zf);

        f4 m = *(const f4*)(mask + (size_t)bb * SSS + idx);

        if (z0 + 4 > SI) {                  // last z-quad: lane 3 is padding
            f4 vld = { 1.0f, 1.0f, 1.0f, 0.0f };
            pen *= vld;
        }
        accNv += pen * m;
    }

    // ---- denominator: sum of the entire mask (float4 streaming) ----
    f4 accDv = zf;
    const f4* m4 = (const f4*)mask;
    for (unsigned i = tid; i < NM4; i += stride)
        accDv += m4[i];

    float accN = accNv[0] + accNv[1] + accNv[2] + accNv[3];
    float accD = accDv[0] + accDv[1] + accDv[2] + accDv[3];

    // all threads converged here -> EXEC all 1s for WMMA
    float wN = wave_sum_wmma(accN);
    float wD = wave_sum_wmma(accD);

    if ((threadIdx.x & 31) == 0) {
        atomicAdd(&ws[0], wN);
        atomicAdd(&ws[1], wD);
    }
}

__global__ void jdet_finalize(const float* __restrict__ ws,
                              float* __restrict__ out) {
    out[0] = ws[0] / (ws[1] + 1e-6f);
}

extern "C" void kernel_launch(void* const* d_in, const int* in_sizes, int n_in,
                              void* d_out, int out_size, void* d_ws, size_t ws_size,
                              hipStream_t stream) {
    const float* disp = (const float*)d_in[0];
    const float* mask = (const float*)d_in[1];
    float* out = (float*)d_out;
    float* ws  = (float*)d_ws;

    jdet_init_ws<<<1, 1, 0, stream>>>(ws);

    // 4096 blocks x 256 threads (8 waves/block) -> ~3.3 z-quads per thread;
    // plenty of waves to cover latency while keeping atomic count at ~32K.
    jdet_main<<<4096, 256, 0, stream>>>(disp, mask, ws);

    jdet_finalize<<<1, 1, 0, stream>>>(ws, out);
}